// HeteroGAT_45552423141799
// MI455X (gfx1250) — compile-verified
//
#include <hip/hip_runtime.h>

// ---------------------------------------------------------------------------
// HeteroGAT forward for MI455X (gfx1250, wave32).
// Dense projections -> v_wmma_f32_16x16x4_f32, 16x64 tile per wave, fully
// unrolled guard-free inner loop over fragment-packed weights.
// Edge phases -> atomics + LDS-accumulated scatter for tiny destination sets.
// Workspace requirement: ~110 MB.
// ---------------------------------------------------------------------------

#define NPAT   100000
#define NSIG   96
#define NCON   32
#define HIDN   64
#define E_TO   1500000
#define E_HAS  500000
#define E_FOL  1600000
#define SLOPE  0.2f

typedef __attribute__((ext_vector_type(2))) float v2f;
typedef __attribute__((ext_vector_type(8))) float v8f;

// ---- helpers ---------------------------------------------------------------

__device__ __forceinline__ unsigned encf(float f) {
  unsigned u = __float_as_uint(f);
  return (u & 0x80000000u) ? ~u : (u | 0x80000000u);   // order-preserving map
}
__device__ __forceinline__ float decf(unsigned k) {
  if (k == 0u) return 0.0f;                            // empty segment -> 0
  unsigned u = (k & 0x80000000u) ? (k ^ 0x80000000u) : ~k;
  float f = __uint_as_float(u);
  if (!isfinite(f)) return 0.0f;
  return f;
}
__device__ __forceinline__ void atomAddF(float* p, float v) {
  __hip_atomic_fetch_add(p, v, __ATOMIC_RELAXED, __HIP_MEMORY_SCOPE_AGENT);
}
__device__ __forceinline__ void atomAddLds(float* p, float v) {
  __hip_atomic_fetch_add(p, v, __ATOMIC_RELAXED, __HIP_MEMORY_SCOPE_WORKGROUP);
}

// ---- fills -----------------------------------------------------------------

__global__ void fill_f32(float* p, float v, int n) {
  int i = blockIdx.x * blockDim.x + threadIdx.x;
  if (i < n) p[i] = v;
}
__global__ void fill_u32(unsigned* p, unsigned v, int n) {
  int i = blockIdx.x * blockDim.x + threadIdx.x;
  if (i < n) p[i] = v;
}

// ---- weight fragment pack --------------------------------------------------
// Wp[(kp*64 + n)*2 + j] = W[2*kp + j][n]  (zero for rows >= K).
// One v2f load then yields the B-fragment pair {W[ka][n], W[ka+1][n]}.

__global__ void packW(const float* __restrict__ W, float* __restrict__ Wp,
                      int Kp, int K) {
  int idx = blockIdx.x * blockDim.x + threadIdx.x;
  int np = (Kp >> 1) * HIDN;
  if (idx >= np) return;
  int kp = idx >> 6, n = idx & 63;
  int r0 = 2 * kp, r1 = 2 * kp + 1;
  float w0 = W[(r0 < K ? r0 : 0) * HIDN + n] * (r0 < K ? 1.0f : 0.0f);
  float w1 = W[(r1 < K ? r1 : 0) * HIDN + n] * (r1 < K ? 1.0f : 0.0f);
  *(float2*)(Wp + (size_t)idx * 2) = make_float2(w0, w1);
}

// ---- WMMA GEMM -------------------------------------------------------------
// Wave computes a 16-row x 64-col tile of out = X[M x KA] @ W[KA x 64] (+bias).
// A 16x4 f32 frag: lanes0-15 M=0..15 K={0,1}; lanes16-31 K={2,3}.
// B frag from packed Wp: one b64 load per tile per k-step.
// C/D: col = lane%16; element r holds row = r + (lane/16)*8.
// A-row index CLAMPED to M-1: garbage only lands in D rows >= M (not stored),
// so all loads are unconditional. KP = KA rounded up to multiple of 4; the
// tail chunk (and odd-aligned KA) uses clamp+mask scalar loads (still
// unconditional -> no exec-mask branching).

__device__ __forceinline__ void gemm_store(const v8f acc[4], const float* bias,
                                           float* __restrict__ out, int M,
                                           int m0, int l16, int hi) {
  if (m0 + 15 < M) {  // fast path: whole row tile in range
#pragma unroll
    for (int t = 0; t < 4; ++t) {
      const int col = t * 16 + l16;
      const float bv = bias ? bias[col] : 0.0f;
#pragma unroll
      for (int r = 0; r < 8; ++r)
        out[(size_t)(m0 + r + hi * 8) * HIDN + col] = acc[t][r] + bv;
    }
  } else {
#pragma unroll
    for (int t = 0; t < 4; ++t) {
      const int col = t * 16 + l16;
      const float bv = bias ? bias[col] : 0.0f;
#pragma unroll
      for (int r = 0; r < 8; ++r) {
        int m = m0 + r + hi * 8;
        if (m < M) out[(size_t)m * HIDN + col] = acc[t][r] + bv;
      }
    }
  }
}

template <int KA, int KP>
__global__ __launch_bounds__(256) void gemm_tile(const float* __restrict__ X,
                                                 const float* __restrict__ Wp,
                                                 const float* __restrict__ bias,
                                                 float* __restrict__ out, int M) {
  const int lane = threadIdx.x & 31;
  const int wv   = threadIdx.x >> 5;
  const int m0   = (blockIdx.x * (blockDim.x >> 5) + wv) * 16;
  if (m0 >= M) return;
  const int l16 = lane & 15;
  const int hi  = lane >> 4;
  int arow = m0 + l16;
  if (arow >= M) arow = M - 1;                 // clamp (see note)
  const float* __restrict__ Xr = X + (size_t)arow * KA;
  const v2f* __restrict__ Bp = ((const v2f*)Wp) + l16;
  v8f acc0 = {}, acc1 = {}, acc2 = {}, acc3 = {};
#pragma unroll
  for (int k0 = 0; k0 < KP; k0 += 4) {
    v2f a;
    if ((k0 + 4 <= KA) && (KA % 2 == 0)) {     // folds per unrolled iteration
      a = *(const v2f*)(Xr + k0 + hi * 2);     // aligned contiguous pair
    } else {                                   // tail / unaligned KA
      const int ka = k0 + hi * 2;
      const int c0 = (ka     < KA) ? ka     : 0;
      const int c1 = (ka + 1 < KA) ? ka + 1 : 0;
      const float m0v = (ka     < KA) ? 1.0f : 0.0f;
      const float m1v = (ka + 1 < KA) ? 1.0f : 0.0f;
      a[0] = Xr[c0] * m0v;                     // unconditional loads
      a[1] = Xr[c1] * m1v;
    }
    const v2f* __restrict__ B = Bp + ((k0 >> 1) + hi) * HIDN;
    acc0 = __builtin_amdgcn_wmma_f32_16x16x4_f32(false, a, false, B[0],  (short)0, acc0, false, false);
    acc1 = __builtin_amdgcn_wmma_f32_16x16x4_f32(false, a, false, B[16], (short)0, acc1, false, false);
    acc2 = __builtin_amdgcn_wmma_f32_16x16x4_f32(false, a, false, B[32], (short)0, acc2, false, false);
    acc3 = __builtin_amdgcn_wmma_f32_16x16x4_f32(false, a, false, B[48], (short)0, acc3, false, false);
  }
  const v8f acc[4] = {acc0, acc1, acc2, acc3};
  gemm_store(acc, bias, out, M, m0, l16, hi);
}

// ---- per-node head scores: a[i,h] = sum_c h[i,h*16+c]*att[h,c] -------------

__global__ void headscore(const float* __restrict__ hm, const float* __restrict__ att,
                          float* __restrict__ out, int n) {
  int i = blockIdx.x * blockDim.x + threadIdx.x;
  if (i >= n) return;
  const float4* row = (const float4*)(hm + (size_t)i * HIDN);
  const float4* av  = (const float4*)att;
  float a[4] = {0.f, 0.f, 0.f, 0.f};
#pragma unroll
  for (int q = 0; q < 16; ++q) {
    float4 v = row[q];
    float4 w = av[q];
    a[q >> 2] += v.x * w.x + v.y * w.y + v.z * w.z + v.w * w.w;
  }
  *(float4*)(out + (size_t)i * 4) = make_float4(a[0], a[1], a[2], a[3]);
}

// ---- fold w_dst (64x64) with att_dst (4x16) -> wfold (64x4) ----------------

__global__ void foldw(const float* __restrict__ w, const float* __restrict__ att,
                      float* __restrict__ wf) {
  int t = threadIdx.x;
  if (t >= 256) return;
  int k = t >> 2, h = t & 3;
  float s = 0.f;
  for (int c = 0; c < 16; ++c) s += w[k * HIDN + h * 16 + c] * att[h * 16 + c];
  wf[k * 4 + h] = s;
}

// ---- folded score: out[i,h] = sum_k x[i,k]*wf[k,h] -------------------------

__global__ void matvec4(const float* __restrict__ x, const float* __restrict__ wf,
                        float* __restrict__ out, int n) {
  int i = blockIdx.x * blockDim.x + threadIdx.x;
  if (i >= n) return;
  const float4* row = (const float4*)(x + (size_t)i * HIDN);
  float a0 = 0.f, a1 = 0.f, a2 = 0.f, a3 = 0.f;
#pragma unroll
  for (int q = 0; q < 16; ++q) {
    float4 v = row[q];
    const float4* wr = (const float4*)(wf + q * 16);
    float4 w0 = wr[0], w1 = wr[1], w2 = wr[2], w3 = wr[3];
    a0 += v.x * w0.x + v.y * w1.x + v.z * w2.x + v.w * w3.x;
    a1 += v.x * w0.y + v.y * w1.y + v.z * w2.y + v.w * w3.y;
    a2 += v.x * w0.z + v.y * w1.z + v.z * w2.z + v.w * w3.z;
    a3 += v.x * w0.w + v.y * w1.w + v.z * w2.w + v.w * w3.w;
  }
  *(float4*)(out + (size_t)i * 4) = make_float4(a0, a1, a2, a3);
}

// ---- edge-scalar constant: ec[h] = sum_c w_edge[h*16+c]*att_edge[h,c] ------

__global__ void econstk(const float* __restrict__ we, const float* __restrict__ ae,
                        float* __restrict__ out) {
  int h = threadIdx.x;
  if (h >= 4) return;
  float s = 0.f;
  for (int c = 0; c < 16; ++c) s += we[h * 16 + c] * ae[h * 16 + c];
  out[h] = s;
}

// ---- GAT pass1: alpha = leakyrelu(a_src[s]+a_dst[d]+ew*ec), seg-max --------

__global__ void gat_pass1(const float* __restrict__ as_, const float* __restrict__ ad_,
                          const int* __restrict__ src, const int* __restrict__ dst,
                          const float* __restrict__ ew, const float* __restrict__ ec,
                          int rev, int E, float* __restrict__ alpha,
                          unsigned* __restrict__ amax) {
  int e = blockIdx.x * blockDim.x + threadIdx.x;
  if (e >= E) return;
  int s = src[e], d = dst[e];
  float4 as4 = *(const float4*)(as_ + (size_t)s * 4);
  float4 ad4 = *(const float4*)(ad_ + (size_t)d * 4);
  float a[4] = {as4.x + ad4.x, as4.y + ad4.y, as4.z + ad4.z, as4.w + ad4.w};
  if (ew) {
    float w = ew[rev ? (E - 1 - e) : e];
    float4 ec4 = *(const float4*)ec;
    a[0] += w * ec4.x; a[1] += w * ec4.y; a[2] += w * ec4.z; a[3] += w * ec4.w;
  }
#pragma unroll
  for (int h = 0; h < 4; ++h) {
    a[h] = (a[h] > 0.0f) ? a[h] : SLOPE * a[h];
    atomicMax(&amax[d * 4 + h], encf(a[h]));
  }
  *(float4*)(alpha + (size_t)e * 4) = make_float4(a[0], a[1], a[2], a[3]);
}

// ---- GAT pass2: ex = exp(alpha - amax[dst]); den = seg-sum -----------------

__global__ void gat_pass2(float* __restrict__ alpha, const unsigned* __restrict__ amax,
                          const int* __restrict__ dst, float* __restrict__ den, int E) {
  int e = blockIdx.x * blockDim.x + threadIdx.x;
  if (e >= E) return;
  int d = dst[e];
  uint4 mk = *(const uint4*)(amax + (size_t)d * 4);
  float4 al = *(const float4*)(alpha + (size_t)e * 4);
  float ex[4];
  ex[0] = __expf(al.x - decf(mk.x));
  ex[1] = __expf(al.y - decf(mk.y));
  ex[2] = __expf(al.z - decf(mk.z));
  ex[3] = __expf(al.w - decf(mk.w));
  *(float4*)(alpha + (size_t)e * 4) = make_float4(ex[0], ex[1], ex[2], ex[3]);
#pragma unroll
  for (int h = 0; h < 4; ++h) atomAddF(&den[d * 4 + h], ex[h]);
}

// ---- GAT pass3 (large dst set): direct global atomics, warp per edge -------

__global__ void gat_scatter_big(const float* __restrict__ hsrc, const float* __restrict__ ex,
                                const float* __restrict__ den, const int* __restrict__ src,
                                const int* __restrict__ dst, int E, float* __restrict__ acc) {
  long long idx = (long long)blockIdx.x * blockDim.x + threadIdx.x;
  long long tot = (long long)E * 32;
  if (idx >= tot) return;
  int e = (int)(idx >> 5), lane = (int)(idx & 31);
  int s = src[e], d = dst[e];
  int c = lane * 2, h = c >> 4;
  float att = ex[(size_t)e * 4 + h] / (den[d * 4 + h] + 1e-16f);
  float2 hv = *(const float2*)(hsrc + (size_t)s * HIDN + c);
  atomAddF(&acc[(size_t)d * HIDN + c],     hv.x * att);
  atomAddF(&acc[(size_t)d * HIDN + c + 1], hv.y * att);
}

// ---- GAT pass3 (tiny dst set: 96/32 nodes): LDS accumulation ---------------

__global__ void gat_scatter_small(const float* __restrict__ hsrc, const float* __restrict__ ex,
                                  const float* __restrict__ den, const int* __restrict__ src,
                                  const int* __restrict__ dst, int E, int ndst,
                                  float* __restrict__ acc) {
  extern __shared__ float sm[];
  int nel = ndst * HIDN;
  for (int i = threadIdx.x; i < nel; i += blockDim.x) sm[i] = 0.0f;
  __syncthreads();
  int lane = threadIdx.x & 31, warp = threadIdx.x >> 5, wpb = blockDim.x >> 5;
  int c = lane * 2, h = c >> 4;
  for (int e = blockIdx.x * wpb + warp; e < E; e += gridDim.x * wpb) {
    int s = src[e], d = dst[e];
    float att = ex[(size_t)e * 4 + h] / (den[d * 4 + h] + 1e-16f);
    float2 hv = *(const float2*)(hsrc + (size_t)s * HIDN + c);
    atomAddLds(&sm[d * HIDN + c],     hv.x * att);
    atomAddLds(&sm[d * HIDN + c + 1], hv.y * att);
  }
  __syncthreads();
  for (int i = threadIdx.x; i < nel; i += blockDim.x) atomAddF(&acc[i], sm[i]);
}

// ---- GCN helpers -----------------------------------------------------------

__global__ void deg_count(const int* __restrict__ dst, int E, float* __restrict__ deg) {
  int e = blockIdx.x * blockDim.x + threadIdx.x;
  if (e < E) atomAddF(&deg[dst[e]], 1.0f);
}
__global__ void rsqrt_k(float* p, int n) {
  int i = blockIdx.x * blockDim.x + threadIdx.x;
  if (i < n) p[i] = rsqrtf(p[i]);
}
__global__ void gcn_scatter(const float* __restrict__ xw, const int* __restrict__ src,
                            const int* __restrict__ dst, const float* __restrict__ dinv,
                            int E, float* __restrict__ acc) {
  long long idx = (long long)blockIdx.x * blockDim.x + threadIdx.x;
  long long tot = (long long)E * 32;
  if (idx >= tot) return;
  int e = (int)(idx >> 5), lane = (int)(idx & 31);
  int s = src[e], d = dst[e];
  float coef = dinv[s] * dinv[d];
  int c = lane * 2;
  float2 v = *(const float2*)(xw + (size_t)s * HIDN + c);
  atomAddF(&acc[(size_t)d * HIDN + c],     v.x * coef);
  atomAddF(&acc[(size_t)d * HIDN + c + 1], v.y * coef);
}
__global__ void gcn_self(const float* __restrict__ xw, const float* __restrict__ dinv,
                         float* __restrict__ acc, int n) {
  int i = blockIdx.x * blockDim.x + threadIdx.x;
  if (i >= n * HIDN) return;
  float dv = dinv[i >> 6];
  acc[i] += xw[i] * dv * dv;
}

// ---- residual + biases + relu ---------------------------------------------

__global__ void combine_relu(const float* __restrict__ acc, float* __restrict__ x,
                             const float* b0, const float* b1,
                             const float* b2, const float* b3, int n) {
  int i = blockIdx.x * blockDim.x + threadIdx.x;
  if (i >= n * HIDN) return;
  int c = i & 63;
  float b = 0.f;
  if (b0) b += b0[c];
  if (b1) b += b1[c];
  if (b2) b += b2[c];
  if (b3) b += b3[c];
  float v = acc[i] + x[i] + b;
  x[i] = v > 0.f ? v : 0.f;
}

// ---------------------------------------------------------------------------
// Host orchestration
// ---------------------------------------------------------------------------

struct LinP { const float* w; const float* b; };
struct GatP {
  const float* w_src; const float* w_dst;
  const float* att_src; const float* att_dst;
  const float* bias; const float* w_edge; const float* att_edge;
};
struct LayerP { GatP to, to_rev, has, has_rev; LinP fol, folrev; };

extern "C" void kernel_launch(void* const* d_in, const int* in_sizes, int n_in,
                              void* d_out, int out_size, void* d_ws, size_t ws_size,
                              hipStream_t stream) {
  (void)in_sizes; (void)n_in; (void)out_size; (void)ws_size;
  // ---- parse inputs (setup_inputs insertion order; params flattened DF) ----
  int ix = 0;
  auto F = [&]() { return (const float*)d_in[ix++]; };
  const float* x_pat  = F();
  const float* x_sig  = F();
  const float* x_con  = F();
  const float* edge_w = F();
  LinP ip_pat{F(), F()};
  LinP ip_con{F(), F()};
  LinP ip_sig{F(), F()};
  LayerP Ls[4];
  for (int l = 0; l < 4; ++l) {
    Ls[l].to      = GatP{F(), F(), F(), F(), F(), F(), F()};
    Ls[l].to_rev  = GatP{F(), F(), F(), F(), F(), F(), F()};
    Ls[l].has     = GatP{F(), F(), F(), F(), F(), nullptr, nullptr};
    Ls[l].has_rev = GatP{F(), F(), F(), F(), F(), nullptr, nullptr};
    Ls[l].fol     = LinP{F(), F()};
    Ls[l].folrev  = LinP{F(), F()};
  }
  LinP op_pat{F(), F()};
  LinP op_con{F(), F()};
  LinP op_sig{F(), F()};
  const int* ei_to  = (const int*)d_in[ix++];
  const int* ei_has = (const int*)d_in[ix++];
  const int* ei_fol = (const int*)d_in[ix++];

  // ---- workspace layout ----------------------------------------------------
  char* base = (char*)d_ws;
  size_t cur = 0;
  auto alloc = [&](size_t bytes) -> void* {
    size_t a = (cur + 255) & ~(size_t)255;
    void* p = base + a;
    cur = a + bytes;
    return p;
  };
  float* xp      = (float*)alloc((size_t)NPAT * HIDN * 4);
  float* acc_p   = (float*)alloc((size_t)NPAT * HIDN * 4);
  float* tmp_h   = (float*)alloc((size_t)NPAT * HIDN * 4);   // hs(p) / xw
  float* alpha   = (float*)alloc((size_t)E_TO * 4 * 4);      // per-edge alpha/ex
  float* sc_src  = (float*)alloc((size_t)NPAT * 4 * 4);
  float* sc_dst  = (float*)alloc((size_t)NPAT * 4 * 4);
  unsigned* amax = (unsigned*)alloc((size_t)NPAT * 4 * 4);
  float* den     = (float*)alloc((size_t)NPAT * 4 * 4);
  float* dinv_f  = (float*)alloc((size_t)NPAT * 4);
  float* dinv_r  = (float*)alloc((size_t)NPAT * 4);
  float* xs      = (float*)alloc((size_t)NSIG * HIDN * 4);
  float* acc_s   = (float*)alloc((size_t)NSIG * HIDN * 4);
  float* sm_h    = (float*)alloc((size_t)NSIG * HIDN * 4);   // small-side h
  float* xc      = (float*)alloc((size_t)NCON * HIDN * 4);
  float* acc_c   = (float*)alloc((size_t)NCON * HIDN * 4);
  float* wp      = (float*)alloc((size_t)140 * HIDN * 4);    // packed weights
  float* wfold   = (float*)alloc(64 * 4 * 4);
  float* ecbuf   = (float*)alloc(4 * 4);

  const int T = 256;
  auto blk = [&](long long n) { return (unsigned)((n + T - 1) / T); };
  // 256-thread blocks = 8 waves = 8 row tiles (128 rows) per block.
  auto gemm = [&](const float* X, const float* W, const float* b, float* o, int M, int K) {
    int Kp = (K + 3) & ~3;
    packW<<<blk((Kp >> 1) * HIDN), T, 0, stream>>>(W, wp, Kp, K);
    dim3 g((M + 127) / 128), t(256);
    switch (K) {
      case 64:  gemm_tile<64, 64><<<g, t, 0, stream>>>(X, wp, b, o, M);   break;
      case 96:  gemm_tile<96, 96><<<g, t, 0, stream>>>(X, wp, b, o, M);   break;
      case 138: gemm_tile<138, 140><<<g, t, 0, stream>>>(X, wp, b, o, M); break;
      case 7:   gemm_tile<7, 8><<<g, t, 0, stream>>>(X, wp, b, o, M);     break;
      default:  break;
    }
  };

  // ---- input projections ---------------------------------------------------
  gemm(x_pat, ip_pat.w, ip_pat.b, xp, NPAT, 138);
  gemm(x_sig, ip_sig.w, ip_sig.b, xs, NSIG, 96);
  gemm(x_con, ip_con.w, ip_con.b, xc, NCON, 7);

  // ---- GCN degree norms (layer-invariant) ----------------------------------
  fill_f32<<<blk(NPAT), T, 0, stream>>>(dinv_f, 1.0f, NPAT);
  deg_count<<<blk(E_FOL), T, 0, stream>>>(ei_fol + E_FOL, E_FOL, dinv_f);
  rsqrt_k<<<blk(NPAT), T, 0, stream>>>(dinv_f, NPAT);
  fill_f32<<<blk(NPAT), T, 0, stream>>>(dinv_r, 1.0f, NPAT);
  deg_count<<<blk(E_FOL), T, 0, stream>>>(ei_fol, E_FOL, dinv_r);
  rsqrt_k<<<blk(NPAT), T, 0, stream>>>(dinv_r, NPAT);

  const int* to_s = ei_to;          const int* to_d = ei_to + E_TO;
  const int* ha_s = ei_has;         const int* ha_d = ei_has + E_HAS;
  const int* fo_s = ei_fol;         const int* fo_d = ei_fol + E_FOL;

  for (int l = 0; l < 4; ++l) {
    const LayerP& L = Ls[l];
    fill_f32<<<blk((size_t)NPAT * HIDN), T, 0, stream>>>(acc_p, 0.f, NPAT * HIDN);
    fill_f32<<<blk(NSIG * HIDN), T, 0, stream>>>(acc_s, 0.f, NSIG * HIDN);
    fill_f32<<<blk(NCON * HIDN), T, 0, stream>>>(acc_c, 0.f, NCON * HIDN);

    // ---- GAT 'to': patient -> signature (96 dst, LDS scatter) ----
    gemm(xp, L.to.w_src, nullptr, tmp_h, NPAT, HIDN);
    headscore<<<blk(NPAT), T, 0, stream>>>(tmp_h, L.to.att_src, sc_src, NPAT);
    gemm(xs, L.to.w_dst, nullptr, sm_h, NSIG, HIDN);
    headscore<<<blk(NSIG), T, 0, stream>>>(sm_h, L.to.att_dst, sc_dst, NSIG);
    econstk<<<1, 64, 0, stream>>>(L.to.w_edge, L.to.att_edge, ecbuf);
    fill_u32<<<blk(NSIG * 4), T, 0, stream>>>(amax, 0u, NSIG * 4);
    fill_f32<<<blk(NSIG * 4), T, 0, stream>>>(den, 0.f, NSIG * 4);
    gat_pass1<<<blk(E_TO), T, 0, stream>>>(sc_src, sc_dst, to_s, to_d, edge_w, ecbuf,
                                           0, E_TO, alpha, amax);
    gat_pass2<<<blk(E_TO), T, 0, stream>>>(alpha, amax, to_d, den, E_TO);
    gat_scatter_small<<<512, T, NSIG * HIDN * 4, stream>>>(tmp_h, alpha, den, to_s, to_d,
                                                           E_TO, NSIG, acc_s);

    // ---- GAT 'to_rev': signature -> patient (folded dst scores) ----
    gemm(xs, L.to_rev.w_src, nullptr, sm_h, NSIG, HIDN);
    headscore<<<blk(NSIG), T, 0, stream>>>(sm_h, L.to_rev.att_src, sc_src, NSIG);
    foldw<<<1, 256, 0, stream>>>(L.to_rev.w_dst, L.to_rev.att_dst, wfold);
    matvec4<<<blk(NPAT), T, 0, stream>>>(xp, wfold, sc_dst, NPAT);
    econstk<<<1, 64, 0, stream>>>(L.to_rev.w_edge, L.to_rev.att_edge, ecbuf);
    fill_u32<<<blk((size_t)NPAT * 4), T, 0, stream>>>(amax, 0u, NPAT * 4);
    fill_f32<<<blk((size_t)NPAT * 4), T, 0, stream>>>(den, 0.f, NPAT * 4);
    gat_pass1<<<blk(E_TO), T, 0, stream>>>(sc_src, sc_dst, to_d, to_s, edge_w, ecbuf,
                                           1, E_TO, alpha, amax);
    gat_pass2<<<blk(E_TO), T, 0, stream>>>(alpha, amax, to_s, den, E_TO);
    gat_scatter_big<<<blk((long long)E_TO * 32), T, 0, stream>>>(sm_h, alpha, den,
                                                                 to_d, to_s, E_TO, acc_p);

    // ---- GAT 'has': patient -> condition (32 dst, LDS scatter) ----
    gemm(xp, L.has.w_src, nullptr, tmp_h, NPAT, HIDN);
    headscore<<<blk(NPAT), T, 0, stream>>>(tmp_h, L.has.att_src, sc_src, NPAT);
    gemm(xc, L.has.w_dst, nullptr, sm_h, NCON, HIDN);
    headscore<<<blk(NCON), T, 0, stream>>>(sm_h, L.has.att_dst, sc_dst, NCON);
    fill_u32<<<blk(NCON * 4), T, 0, stream>>>(amax, 0u, NCON * 4);
    fill_f32<<<blk(NCON * 4), T, 0, stream>>>(den, 0.f, NCON * 4);
    gat_pass1<<<blk(E_HAS), T, 0, stream>>>(sc_src, sc_dst, ha_s, ha_d, nullptr, nullptr,
                                            0, E_HAS, alpha, amax);
    gat_pass2<<<blk(E_HAS), T, 0, stream>>>(alpha, amax, ha_d, den, E_HAS);
    gat_scatter_small<<<512, T, NCON * HIDN * 4, stream>>>(tmp_h, alpha, den, ha_s, ha_d,
                                                           E_HAS, NCON, acc_c);

    // ---- GAT 'has_rev': condition -> patient ----
    gemm(xc, L.has_rev.w_src, nullptr, sm_h, NCON, HIDN);
    headscore<<<blk(NCON), T, 0, stream>>>(sm_h, L.has_rev.att_src, sc_src, NCON);
    foldw<<<1, 256, 0, stream>>>(L.has_rev.w_dst, L.has_rev.att_dst, wfold);
    matvec4<<<blk(NPAT), T, 0, stream>>>(xp, wfold, sc_dst, NPAT);
    fill_u32<<<blk((size_t)NPAT * 4), T, 0, stream>>>(amax, 0u, NPAT * 4);
    fill_f32<<<blk((size_t)NPAT * 4), T, 0, stream>>>(den, 0.f, NPAT * 4);
    gat_pass1<<<blk(E_HAS), T, 0, stream>>>(sc_src, sc_dst, ha_d, ha_s, nullptr, nullptr,
                                            0, E_HAS, alpha, amax);
    gat_pass2<<<blk(E_HAS), T, 0, stream>>>(alpha, amax, ha_s, den, E_HAS);
    gat_scatter_big<<<blk((long long)E_HAS * 32), T, 0, stream>>>(sm_h, alpha, den,
                                                                  ha_d, ha_s, E_HAS, acc_p);

    // ---- GCN follows (src=row0, dst=row1, dinv_f) ----
    gemm(xp, L.fol.w, nullptr, tmp_h, NPAT, HIDN);
    gcn_scatter<<<blk((long long)E_FOL * 32), T, 0, stream>>>(tmp_h, fo_s, fo_d, dinv_f,
                                                              E_FOL, acc_p);
    gcn_self<<<blk((size_t)NPAT * HIDN), T, 0, stream>>>(tmp_h, dinv_f, acc_p, NPAT);

    // ---- GCN follows_rev (src=row1, dst=row0, dinv_r) ----
    gemm(xp, L.folrev.w, nullptr, tmp_h, NPAT, HIDN);
    gcn_scatter<<<blk((long long)E_FOL * 32), T, 0, stream>>>(tmp_h, fo_d, fo_s, dinv_r,
                                                              E_FOL, acc_p);
    gcn_self<<<blk((size_t)NPAT * HIDN), T, 0, stream>>>(tmp_h, dinv_r, acc_p, NPAT);

    // ---- residual + biases + relu ----
    combine_relu<<<blk((size_t)NPAT * HIDN), T, 0, stream>>>(
        acc_p, xp, L.to_rev.bias, L.has_rev.bias, L.fol.b, L.folrev.b, NPAT);
    combine_relu<<<blk(NSIG * HIDN), T, 0, stream>>>(
        acc_s, xs, L.to.bias, nullptr, nullptr, nullptr, NSIG);
    combine_relu<<<blk(NCON * HIDN), T, 0, stream>>>(
        acc_c, xc, L.has.bias, nullptr, nullptr, nullptr, NCON);
  }

  // ---- output projections --------------------------------------------------
  float* outp = (float*)d_out;
  float* outs = outp + (size_t)NPAT * HIDN;
  float* outc = outs + (size_t)NSIG * HIDN;
  gemm(xp, op_pat.w, op_pat.b, outp, NPAT, HIDN);
  gemm(xs, op_sig.w, op_sig.b, outs, NSIG, HIDN);
  gemm(xc, op_con.w, op_con.b, outc, NCON, HIDN);
}